// QRDcell2_34557306864069
// MI455X (gfx1250) — compile-verified
//
#include <hip/hip_runtime.h>

// CDNA5 / gfx1250, wave32. One workgroup per batch chain; f32 WMMA 16x16x4.
// Round 2: software-pipelined matmul inner loop (B/A frags for kk+1 staged in
// registers before the WMMA burst for kk) so LDS/global loads overlap WMMA.

typedef __attribute__((ext_vector_type(2))) float v2f;
typedef __attribute__((ext_vector_type(8))) float v8f;

#define DMAT 128
#define TLEN 256
#define CSTR 132   // padded column stride (floats) to dodge LDS bank conflicts

__global__ __launch_bounds__(256) void qr_prop_kernel(const float* __restrict__ J,
                                                      float* __restrict__ out) {
  extern __shared__ float Mc[];   // column-major: col j at Mc + j*CSTR, 128 rows used
  __shared__ float red[8];

  const int tid  = threadIdx.x;
  const int wid  = tid >> 5;      // 8 waves
  const int lane = tid & 31;
  const int l16  = lane & 15;
  const int half = lane >> 4;     // 0 or 1
  const int b    = blockIdx.x;

  // Q <- I  (column-major, pad rows zeroed)
  for (int idx = tid; idx < DMAT * CSTR; idx += 256) {
    const int c = idx / CSTR;
    const int r = idx - c * CSTR;
    Mc[idx] = (r == c) ? 1.0f : 0.0f;
  }
  __syncthreads();

  const int m0   = wid * 16;          // wave's C row-block
  const int arow = m0 + l16;          // A-frag row for this lane
  const int koff = 2 * half;          // A/B-frag K sub-offset for this half-wave
  const float* bbase0 = Mc + l16 * CSTR + koff;   // B-frag base for this lane
  float* outb = out + (size_t)b * TLEN * DMAT;

  for (int t = 0; t < TLEN; ++t) {
    const float* Jt = J + (((size_t)b * TLEN + t) << 14);   // 128*128 = 16384
    const float* abase = Jt + arow * DMAT + koff;

    // Prefetch next step's 64KB J tile into the hierarchy (global_prefetch_b8).
    if (t + 1 < TLEN) {
      const char* nj = (const char*)(Jt + (1 << 14));
      __builtin_prefetch(nj + tid * 256,       0, 1);
      __builtin_prefetch(nj + tid * 256 + 128, 0, 1);
    }

    // ---------------- M = J_t @ Q : full C held in registers ----------------
    v8f acc[8] = {};

    // Prologue: stage frags for kk = 0
    v2f a = *(const v2f*)(abase);
    v2f bb[8];
#pragma unroll
    for (int nt = 0; nt < 8; ++nt)
      bb[nt] = *(const v2f*)(bbase0 + nt * (16 * CSTR));

    // Steady state: issue kk+1 loads, then kk's 8 WMMAs (loads hide under WMMA)
#pragma unroll 2
    for (int kk = 0; kk < 31; ++kk) {
      v2f an = *(const v2f*)(abase + (kk + 1) * 4);
      v2f bn[8];
#pragma unroll
      for (int nt = 0; nt < 8; ++nt)
        bn[nt] = *(const v2f*)(bbase0 + (kk + 1) * 4 + nt * (16 * CSTR));
#pragma unroll
      for (int nt = 0; nt < 8; ++nt)
        acc[nt] = __builtin_amdgcn_wmma_f32_16x16x4_f32(
            false, a, false, bb[nt], (short)0, acc[nt], false, false);
      a = an;
#pragma unroll
      for (int nt = 0; nt < 8; ++nt) bb[nt] = bn[nt];
    }
    // Epilogue: kk = 31
#pragma unroll
    for (int nt = 0; nt < 8; ++nt)
      acc[nt] = __builtin_amdgcn_wmma_f32_16x16x4_f32(
          false, a, false, bb[nt], (short)0, acc[nt], false, false);

    __syncthreads();   // all reads of Q complete before overwrite

    // Scatter C tiles into Mc (becomes M, column-major)
#pragma unroll
    for (int nt = 0; nt < 8; ++nt) {
#pragma unroll
      for (int r = 0; r < 8; ++r) {
        Mc[(nt * 16 + l16) * CSTR + m0 + 8 * half + r] = acc[nt][r];
      }
    }
    __syncthreads();

    // ------------- in-place right-looking MGS QR; emit diag(R) -------------
    for (int i = 0; i < DMAT; ++i) {
      // norm of column i (threads 0..127 hold one element each)
      float v = 0.0f, p = 0.0f;
      if (tid < DMAT) { v = Mc[i * CSTR + tid]; p = v * v; }
#pragma unroll
      for (int off = 16; off; off >>= 1) p += __shfl_xor(p, off, 32);
      if (lane == 0) red[wid] = p;
      __syncthreads();
      const float s   = red[0] + red[1] + red[2] + red[3] +
                        red[4] + red[5] + red[6] + red[7];
      const float inv = rsqrtf(s);
      if (tid == 0) outb[t * DMAT + i] = s * inv;     // r_ii = sqrt(s)
      if (tid < DMAT) Mc[i * CSTR + tid] = v * inv;   // q_i = v / r_ii
      __syncthreads();

      // rank-1 update of trailing columns, j distributed across the 8 waves
      const float q0 = Mc[i * CSTR + lane];
      const float q1 = Mc[i * CSTR + lane + 32];
      const float q2 = Mc[i * CSTR + lane + 64];
      const float q3 = Mc[i * CSTR + lane + 96];
      for (int j = i + 1 + wid; j < DMAT; j += 8) {
        float* col = Mc + j * CSTR;
        float x0 = col[lane], x1 = col[lane + 32];
        float x2 = col[lane + 64], x3 = col[lane + 96];
        float d = x0 * q0 + x1 * q1 + x2 * q2 + x3 * q3;
#pragma unroll
        for (int off = 16; off; off >>= 1) d += __shfl_xor(d, off, 32);
        col[lane]      = x0 - d * q0;
        col[lane + 32] = x1 - d * q1;
        col[lane + 64] = x2 - d * q2;
        col[lane + 96] = x3 - d * q3;
      }
      __syncthreads();
    }
  }
}

extern "C" void kernel_launch(void* const* d_in, const int* in_sizes, int n_in,
                              void* d_out, int out_size, void* d_ws, size_t ws_size,
                              hipStream_t stream) {
  (void)in_sizes; (void)n_in; (void)out_size; (void)d_ws; (void)ws_size;
  const float* J = (const float*)d_in[0];
  float* out = (float*)d_out;
  const size_t shmem = (size_t)DMAT * CSTR * sizeof(float);  // 67,584 B dynamic LDS
  qr_prop_kernel<<<dim3(16), dim3(256), shmem, stream>>>(J, out);
}